// RGCN_77996606095717
// MI455X (gfx1250) — compile-verified
//
#include <hip/hip_runtime.h>
#include <math.h>

#define NN 50000   // nodes
#define RR 46      // relations
#define BB 30      // bases
#define HH 16      // hidden
#define CC 8       // classes

typedef __attribute__((ext_vector_type(2))) float v2f;
typedef __attribute__((ext_vector_type(8))) float v8f;

__device__ __forceinline__ void atomAddF(float* p, float v) {
  (void)__hip_atomic_fetch_add(p, v, __ATOMIC_RELAXED, __HIP_MEMORY_SCOPE_AGENT);
}

__global__ void k_zero(float* __restrict__ p, int n) {
  int i = blockIdx.x * blockDim.x + threadIdx.x;
  if (i < n) p[i] = 0.0f;
}

// counts[r*N + dst] += 1 per edge
__global__ void k_count(const int* __restrict__ dst, const int* __restrict__ etype,
                        float* __restrict__ counts, int E) {
  int e = blockIdx.x * blockDim.x + threadIdx.x;
  if (e >= E) return;
  atomAddF(&counts[etype[e] * NN + dst[e]], 1.0f);
}

// conv1 edge scatter: hacc[dst] += (comp1[rel] . basis1[:,src,:]) / counts[rel,dst]
__global__ void k_conv1(const int* __restrict__ src, const int* __restrict__ dst,
                        const int* __restrict__ etype,
                        const float* __restrict__ basis1, const float* __restrict__ comp1,
                        const float* __restrict__ counts, float* __restrict__ hacc, int E) {
  __shared__ float sc[RR * BB];
  for (int i = threadIdx.x; i < RR * BB; i += blockDim.x) sc[i] = comp1[i];
  __syncthreads();
  int e = blockIdx.x * blockDim.x + threadIdx.x;
  if (e >= E) return;
  int s = src[e], d = dst[e], r = etype[e];
  const float* crow = &sc[r * BB];
  float acc[HH];
#pragma unroll
  for (int j = 0; j < HH; j++) acc[j] = 0.0f;
  const float* bp = basis1 + (size_t)s * HH;
  for (int b = 0; b < BB; b++) {
    float cb = crow[b];
    const float4* q = reinterpret_cast<const float4*>(bp + (size_t)b * NN * HH);
    float4 q0 = q[0], q1 = q[1], q2 = q[2], q3 = q[3];
    acc[0]  = fmaf(cb, q0.x, acc[0]);  acc[1]  = fmaf(cb, q0.y, acc[1]);
    acc[2]  = fmaf(cb, q0.z, acc[2]);  acc[3]  = fmaf(cb, q0.w, acc[3]);
    acc[4]  = fmaf(cb, q1.x, acc[4]);  acc[5]  = fmaf(cb, q1.y, acc[5]);
    acc[6]  = fmaf(cb, q1.z, acc[6]);  acc[7]  = fmaf(cb, q1.w, acc[7]);
    acc[8]  = fmaf(cb, q2.x, acc[8]);  acc[9]  = fmaf(cb, q2.y, acc[9]);
    acc[10] = fmaf(cb, q2.z, acc[10]); acc[11] = fmaf(cb, q2.w, acc[11]);
    acc[12] = fmaf(cb, q3.x, acc[12]); acc[13] = fmaf(cb, q3.y, acc[13]);
    acc[14] = fmaf(cb, q3.z, acc[14]); acc[15] = fmaf(cb, q3.w, acc[15]);
  }
  float inv = 1.0f / counts[r * NN + d];
  float* hp = hacc + (size_t)d * HH;
#pragma unroll
  for (int j = 0; j < HH; j++) atomAddF(&hp[j], acc[j] * inv);
}

// h = relu(hacc + root1 + bias1), in place
__global__ void k_hfinish(float* __restrict__ h, const float* __restrict__ root1,
                          const float* __restrict__ bias1) {
  int i = blockIdx.x * blockDim.x + threadIdx.x;
  if (i >= NN * HH) return;
  float v = h[i] + root1[i] + bias1[i & (HH - 1)];
  h[i] = v > 0.0f ? v : 0.0f;
}

// w2[r,h,c] = sum_b comp2[r,b] * basis2[b,h,c]   (tiny)
__global__ void k_w2(const float* __restrict__ comp2, const float* __restrict__ basis2,
                     float* __restrict__ w2) {
  int i = blockIdx.x * blockDim.x + threadIdx.x;
  if (i >= RR * HH * CC) return;
  int r = i / (HH * CC);
  int hc = i % (HH * CC);
  float s = 0.0f;
  for (int b = 0; b < BB; b++) s = fmaf(comp2[r * BB + b], basis2[b * HH * CC + hc], s);
  w2[i] = s;
}

// conv2 edge scatter: out[dst] += (h[src] @ w2[rel]) / counts[rel,dst]
__global__ void k_conv2(const int* __restrict__ src, const int* __restrict__ dst,
                        const int* __restrict__ etype,
                        const float* __restrict__ h, const float* __restrict__ w2,
                        const float* __restrict__ counts, float* __restrict__ out, int E) {
  __shared__ float sw[RR * HH * CC];  // 23.5 KB
  for (int i = threadIdx.x; i < RR * HH * CC; i += blockDim.x) sw[i] = w2[i];
  __syncthreads();
  int e = blockIdx.x * blockDim.x + threadIdx.x;
  if (e >= E) return;
  int s = src[e], d = dst[e], r = etype[e];
  float inv = 1.0f / counts[r * NN + d];
  const float4* hq = reinterpret_cast<const float4*>(h + (size_t)s * HH);
  float4 h0 = hq[0], h1 = hq[1], h2 = hq[2], h3 = hq[3];
  float hv[HH] = {h0.x, h0.y, h0.z, h0.w, h1.x, h1.y, h1.z, h1.w,
                  h2.x, h2.y, h2.z, h2.w, h3.x, h3.y, h3.z, h3.w};
  const float* w = &sw[r * HH * CC];
  float o[CC];
#pragma unroll
  for (int c = 0; c < CC; c++) o[c] = 0.0f;
#pragma unroll
  for (int hh = 0; hh < HH; hh++) {
    float x = hv[hh];
#pragma unroll
    for (int c = 0; c < CC; c++) o[c] = fmaf(x, w[hh * CC + c], o[c]);
  }
#pragma unroll
  for (int c = 0; c < CC; c++) atomAddF(&out[(size_t)d * CC + c], o[c] * inv);
}

// out[n,c] += (h @ root2)[n,c] + bias2[c]  via V_WMMA_F32_16X16X4_F32, K=16 in 4 steps.
// B-fragment loads are unconditional (clamped column) with a multiply mask for the
// 8 padding columns, so no exec-mask branches appear between the WMMAs.
__global__ void k_root2_wmma(const float* __restrict__ h, const float* __restrict__ root2,
                             const float* __restrict__ bias2, float* __restrict__ out) {
  int lane = threadIdx.x & 31;
  int wave = threadIdx.x >> 5;
  int m0 = (blockIdx.x * (blockDim.x >> 5) + wave) * 16;  // wave-uniform
  if (m0 >= NN) return;                                    // wave-uniform exit; EXEC all-1s
  int half = lane >> 4;  // 0: lanes 0-15, 1: lanes 16-31
  int l = lane & 15;
  int lc = l < CC ? l : CC - 1;              // clamped (always-valid) column
  float msk = l < CC ? 1.0f : 0.0f;          // zero out the 8 padding columns
  v8f c = {};
#pragma unroll
  for (int k = 0; k < HH; k += 4) {
    // A 16x4 layout: lanes 0-15 hold K=k,k+1; lanes 16-31 hold K=k+2,k+3 (ISA 7.12.2)
    int rowA = m0 + l;
    int rA = rowA < NN ? rowA : NN - 1;  // clamp loads; stores masked below
    int ka = k + half * 2;
    v2f a;
    a.x = h[(size_t)rA * HH + ka];
    a.y = h[(size_t)rA * HH + ka + 1];
    // B 4x16 layout: lanes 0-15 hold rows K=k,k+1; lanes 16-31 rows K=k+2,k+3
    v2f b;
    b.x = root2[(ka + 0) * CC + lc] * msk;
    b.y = root2[(ka + 1) * CC + lc] * msk;
    c = __builtin_amdgcn_wmma_f32_16x16x4_f32(false, a, false, b, (short)0, c,
                                              false, false);
  }
  // C/D layout: VGPR v -> row m0 + v + half*8, column = l
  if (l < CC) {
    float bz = bias2[l];
#pragma unroll
    for (int v = 0; v < 8; v++) {
      int row = m0 + v + half * 8;
      if (row < NN) {
        size_t idx = (size_t)row * CC + l;
        out[idx] = out[idx] + c[v] + bz;
      }
    }
  }
}

// row-wise log_softmax over 8 classes, in place
__global__ void k_lsm(float* __restrict__ out) {
  int n = blockIdx.x * blockDim.x + threadIdx.x;
  if (n >= NN) return;
  float v[CC];
  float m = -INFINITY;
#pragma unroll
  for (int c = 0; c < CC; c++) {
    v[c] = out[(size_t)n * CC + c];
    m = fmaxf(m, v[c]);
  }
  float s = 0.0f;
#pragma unroll
  for (int c = 0; c < CC; c++) s += expf(v[c] - m);
  float lse = m + logf(s);
#pragma unroll
  for (int c = 0; c < CC; c++) out[(size_t)n * CC + c] = v[c] - lse;
}

extern "C" void kernel_launch(void* const* d_in, const int* in_sizes, int n_in,
                              void* d_out, int out_size, void* d_ws, size_t ws_size,
                              hipStream_t stream) {
  const int* edge_index = (const int*)d_in[0];  // [2,E]
  const int* etype      = (const int*)d_in[1];  // [E]
  const float* basis1   = (const float*)d_in[2];  // [B,N,H]
  const float* comp1    = (const float*)d_in[3];  // [R,B]
  const float* root1    = (const float*)d_in[4];  // [N,H]
  const float* bias1    = (const float*)d_in[5];  // [H]
  const float* basis2   = (const float*)d_in[6];  // [B,H,C]
  const float* comp2    = (const float*)d_in[7];  // [R,B]
  const float* root2    = (const float*)d_in[8];  // [H,C]
  const float* bias2    = (const float*)d_in[9];  // [C]
  float* out = (float*)d_out;                     // [N,C] f32

  int E = in_sizes[1];
  const int* src = edge_index;
  const int* dst = edge_index + E;

  float* ws     = (float*)d_ws;
  float* counts = ws;                             // R*N
  float* hbuf   = counts + (size_t)RR * NN;       // N*H
  float* w2     = hbuf + (size_t)NN * HH;         // R*H*C

  const int T = 256;
  k_zero<<<(RR * NN + T - 1) / T, T, 0, stream>>>(counts, RR * NN);
  k_zero<<<(NN * HH + T - 1) / T, T, 0, stream>>>(hbuf, NN * HH);
  k_zero<<<(NN * CC + T - 1) / T, T, 0, stream>>>(out, NN * CC);

  k_count<<<(E + T - 1) / T, T, 0, stream>>>(dst, etype, counts, E);
  k_conv1<<<(E + T - 1) / T, T, 0, stream>>>(src, dst, etype, basis1, comp1, counts,
                                             hbuf, E);
  k_hfinish<<<(NN * HH + T - 1) / T, T, 0, stream>>>(hbuf, root1, bias1);
  k_w2<<<(RR * HH * CC + T - 1) / T, T, 0, stream>>>(comp2, basis2, w2);
  k_conv2<<<(E + T - 1) / T, T, 0, stream>>>(src, dst, etype, hbuf, w2, counts, out, E);
  // 8 waves/block -> 128 rows/block
  k_root2_wmma<<<(NN + 127) / 128, 256, 0, stream>>>(hbuf, root2, bias2, out);
  k_lsm<<<(NN + T - 1) / T, T, 0, stream>>>(out);
}